// VectorQuantizer_20770461844007
// MI455X (gfx1250) — compile-verified
//
#include <hip/hip_runtime.h>
#include <math.h>

typedef __attribute__((ext_vector_type(2))) float v2f;
typedef __attribute__((ext_vector_type(8))) float v8f;

#define NPIX 131072          // N*H*W = 32*64*64
#define NTIL (NPIX / 16)     // 8192 pixel tiles
#define KCB  512             // codebook size

// ---------------------------------------------------------------- utilities
__global__ void zero_accum_kernel(double* __restrict__ acc) {
  int t = threadIdx.x;
  if (t < 34) acc[t] = 0.0;  // [0..15] bnSum, [16..31] bnSumSq, [32] mse1, [33] mse2
}

// ------------------------------------------------ pass 0: |e_n|^2 table (512 floats)
__global__ void e2_kernel(const float* __restrict__ emb, float* __restrict__ e2tab) {
  int n = blockIdx.x * 64 + threadIdx.x;   // 0..511
  const float4* ep = (const float4*)(emb + (size_t)n * 16);
  float s = 0.f;
  #pragma unroll
  for (int q = 0; q < 4; ++q) {
    float4 v = ep[q];
    s += v.x * v.x + v.y * v.y + v.z * v.z + v.w * v.w;
  }
  e2tab[n] = s;
}

// ------------------------------------------------ pass 1: membrane time-collapse
// x [N,C,H,W,T] -> x_mem [pixel(NHW), C]  (NHWC), coef = 0.8^(15-t)
__global__ void memout_kernel(const float* __restrict__ x, float* __restrict__ xmem) {
  int pix = blockIdx.x * 256 + threadIdx.x;   // n*4096 + h*64 + w
  float coef[16];
  coef[15] = 1.f;
  #pragma unroll
  for (int t = 14; t >= 0; --t) coef[t] = coef[t + 1] * 0.8f;
  int n = pix >> 12, hw = pix & 4095;
  float res[16];
  #pragma unroll
  for (int c = 0; c < 16; ++c) {
    const float4* xp = (const float4*)(x + (((size_t)n * 16 + c) * 4096 + hw) * 16);
    float a = 0.f;
    #pragma unroll
    for (int q = 0; q < 4; ++q) {
      float4 v = xp[q];
      a += v.x * coef[4 * q + 0] + v.y * coef[4 * q + 1] +
           v.z * coef[4 * q + 2] + v.w * coef[4 * q + 3];
    }
    res[c] = a;
  }
  float4* op = (float4*)(xmem + (size_t)pix * 16);
  #pragma unroll
  for (int q = 0; q < 4; ++q)
    op[q] = make_float4(res[4 * q + 0], res[4 * q + 1], res[4 * q + 2], res[4 * q + 3]);
}

// ------------------------------------------ pass 2: VQ (WMMA) + 1x1 conv (WMMA)
// one wave == one 16-pixel tile. F32 WMMA 16x16x4 chained over K=16.
__global__ void vq_conv_kernel(const float* __restrict__ xmem,
                               const float* __restrict__ emb,
                               const float* __restrict__ e2tab,
                               const float* __restrict__ convw,
                               const float* __restrict__ convb,
                               float* __restrict__ ybuf,
                               double* __restrict__ bnSum,
                               double* __restrict__ bnSqr,
                               double* __restrict__ mse1Acc) {
  __shared__ int   qidx[4][16];
  __shared__ float bs1[4][16];
  __shared__ float bs2[4][16];
  __shared__ float m1p[4];

  const int wv   = threadIdx.x >> 5;
  const int lane = threadIdx.x & 31;
  const int col  = lane & 15;   // A: pixel row M; B/D: column N; also BN channel o
  const int half = lane >> 4;
  const int tile = blockIdx.x * 4 + wv;

  // A fragment (f32 16x4 layout: lane row = lane%16, K = 2*half + v), 4 K-steps
  const float* fr = xmem + ((size_t)tile * 16 + col) * 16;
  v2f a[4];
  #pragma unroll
  for (int s = 0; s < 4; ++s) {
    a[s][0] = fr[4 * s + 2 * half + 0];
    a[s][1] = fr[4 * s + 2 * half + 1];
  }

  float best[8]; int bidx[8];
  #pragma unroll
  for (int r = 0; r < 8; ++r) { best[r] = 3.4e38f; bidx[r] = 0; }

  // distance GEMM over 32 codeword tiles; score = |e|^2 - 2*f.e (|f|^2 constant in argmin)
  for (int nt = 0; nt < KCB / 16; ++nt) {
    const int n = nt * 16 + col;
    const float* er = emb + (size_t)n * 16;
    v2f b[4];
    #pragma unroll
    for (int s = 0; s < 4; ++s) {
      b[s][0] = er[4 * s + 2 * half + 0];
      b[s][1] = er[4 * s + 2 * half + 1];
    }
    float e2 = e2tab[n];                         // |e_n|^2, precomputed (no DS traffic)
    v8f dacc = {0.f, 0.f, 0.f, 0.f, 0.f, 0.f, 0.f, 0.f};
    #pragma unroll
    for (int s = 0; s < 4; ++s)
      dacc = __builtin_amdgcn_wmma_f32_16x16x4_f32(false, a[s], false, b[s],
                                                   (short)0, dacc, false, false);
    #pragma unroll
    for (int r = 0; r < 8; ++r) {
      float sc = e2 - 2.f * dacc[r];
      if (sc < best[r]) { best[r] = sc; bidx[r] = n; }
    }
  }
  // argmin butterfly across the 16 lanes of each half (tie -> lowest index)
  #pragma unroll
  for (int r = 0; r < 8; ++r) {
    #pragma unroll
    for (int off = 1; off < 16; off <<= 1) {
      float os = __shfl_xor(best[r], off, 32);
      int   oi = __shfl_xor(bidx[r], off, 32);
      if (os < best[r] || (os == best[r] && oi < bidx[r])) { best[r] = os; bidx[r] = oi; }
    }
  }
  if (col == 0) {
    #pragma unroll
    for (int r = 0; r < 8; ++r) qidx[wv][r + 8 * half] = bidx[r];
  }
  __syncthreads();

  // gather codeword rows (A frag of conv GEMM) + VQ mse accumulation
  const int ci = qidx[wv][col];
  const float* qr = emb + (size_t)ci * 16;
  v2f qa[4];
  float m1 = 0.f;
  #pragma unroll
  for (int s = 0; s < 4; ++s) {
    float q0 = qr[4 * s + 2 * half + 0];
    float q1 = qr[4 * s + 2 * half + 1];
    qa[s][0] = q0; qa[s][1] = q1;
    float d0 = q0 - a[s][0], d1 = q1 - a[s][1];
    m1 += d0 * d0 + d1 * d1;   // each lane covers a unique (row, k-half) slice
  }
  // B frag: B[k=c][n=o] = conv_w[o][c], o = col (contiguous in k)
  const float* wr = convw + col * 16;
  v2f wb[4];
  #pragma unroll
  for (int s = 0; s < 4; ++s) {
    wb[s][0] = wr[4 * s + 2 * half + 0];
    wb[s][1] = wr[4 * s + 2 * half + 1];
  }
  v8f yac = {0.f, 0.f, 0.f, 0.f, 0.f, 0.f, 0.f, 0.f};
  #pragma unroll
  for (int s = 0; s < 4; ++s)
    yac = __builtin_amdgcn_wmma_f32_16x16x4_f32(false, qa[s], false, wb[s],
                                                (short)0, yac, false, false);
  const float bo = convb[col];
  float s1 = 0.f, s2 = 0.f;
  #pragma unroll
  for (int r = 0; r < 8; ++r) {
    float yv = yac[r] + bo;                              // y[row r+8*half][o=col]
    s1 += yv; s2 += yv * yv;
    ybuf[((size_t)tile * 16 + r + 8 * half) * 16 + col] = yv;
  }
  s1 += __shfl_xor(s1, 16, 32);    // fold both row-halves; channel o = col
  s2 += __shfl_xor(s2, 16, 32);
  #pragma unroll
  for (int off = 1; off < 32; off <<= 1) m1 += __shfl_xor(m1, off, 32);
  if (half == 0) { bs1[wv][col] = s1; bs2[wv][col] = s2; }
  if (lane == 0) m1p[wv] = m1;
  __syncthreads();
  if (threadIdx.x < 16) {
    float t1 = bs1[0][threadIdx.x] + bs1[1][threadIdx.x] + bs1[2][threadIdx.x] + bs1[3][threadIdx.x];
    float t2 = bs2[0][threadIdx.x] + bs2[1][threadIdx.x] + bs2[2][threadIdx.x] + bs2[3][threadIdx.x];
    atomicAdd(&bnSum[threadIdx.x], (double)t1);
    atomicAdd(&bnSqr[threadIdx.x], (double)t2);
  } else if (threadIdx.x == 16) {
    atomicAdd(mse1Acc, (double)(m1p[0] + m1p[1] + m1p[2] + m1p[3]));
  }
}

// --------------------------------- pass 3: fold BN stats into scale/shift per channel
__global__ void bn_final_kernel(const double* __restrict__ acc,
                                const float* __restrict__ bng,
                                const float* __restrict__ bnb,
                                float* __restrict__ scsh) {
  int t = threadIdx.x;
  if (t < 16) {
    double inv  = 1.0 / (double)NPIX;
    double mean = acc[t] * inv;
    double var  = acc[16 + t] * inv - mean * mean;
    double sc   = 0.2 * (double)bng[t] / sqrt(var + 1e-5);  // ALPHA*VTH = 0.2
    scsh[t]      = (float)sc;
    scsh[16 + t] = (float)((double)bnb[t] - mean * sc);
  }
}

// ------------------ pass 4: normalize + LIF(16 steps) -> spikes p, plus mse2 vs x
__global__ void lif_kernel(const float* __restrict__ ybuf,
                           const float* __restrict__ x,
                           const float* __restrict__ scsh,
                           float* __restrict__ out,
                           double* __restrict__ mse2Acc) {
  __shared__ float red[8];
  size_t tid = (size_t)blockIdx.x * 256 + threadIdx.x;        // (n,c,h,w), w fastest
  int c = (int)((tid >> 12) & 15);
  size_t pixel = (tid >> 16) * 4096 + (tid & 4095);
  float inp = ybuf[pixel * 16 + c] * scsh[c] + scsh[16 + c];  // constant over T

  const float4* xp = (const float4*)(x + tid * 16);
  float4* op = (float4*)(out + tid * 16);
  float u = 0.f, o = 0.f, acc = 0.f;
  #pragma unroll
  for (int q = 0; q < 4; ++q) {
    float4 xv = xp[q];
    float ov[4];
    #pragma unroll
    for (int j = 0; j < 4; ++j) {
      u = 0.25f * u * (1.f - o) + inp;   // TAU * u * (1 - o_prev) + i
      o = (u > 0.2f) ? 1.f : 0.f;        // spike(u), VTH = 0.2
      ov[j] = o;
    }
    float d0 = ov[0] - xv.x, d1 = ov[1] - xv.y, d2 = ov[2] - xv.z, d3 = ov[3] - xv.w;
    acc += d0 * d0 + d1 * d1 + d2 * d2 + d3 * d3;
    op[q] = make_float4(ov[0], ov[1], ov[2], ov[3]);
  }
  #pragma unroll
  for (int off = 1; off < 32; off <<= 1) acc += __shfl_xor(acc, off, 32);
  int wv = threadIdx.x >> 5;
  if ((threadIdx.x & 31) == 0) red[wv] = acc;
  __syncthreads();
  if (threadIdx.x == 0) {
    float s = 0.f;
    #pragma unroll
    for (int i = 0; i < 8; ++i) s += red[i];
    atomicAdd(mse2Acc, (double)s);
  }
}

// --------------------------------------------------- pass 5: combine scalar loss
__global__ void loss_kernel(const double* __restrict__ acc, float* __restrict__ lossOut) {
  double mse1 = acc[32] / (double)(NPIX * 16);                 // 2,097,152 elems
  double mse2 = acc[33] / (double)((size_t)NPIX * 16 * 16);    // 33,554,432 elems
  // loss1 + loss2 = (1+beta)*mse1 + (mse2 + beta*mse1) = 1.5*mse1 + mse2
  lossOut[0] = (float)(1.5 * mse1 + mse2);
}

extern "C" void kernel_launch(void* const* d_in, const int* in_sizes, int n_in,
                              void* d_out, int out_size, void* d_ws, size_t ws_size,
                              hipStream_t stream) {
  const float* x     = (const float*)d_in[0];
  const float* emb   = (const float*)d_in[1];
  const float* convw = (const float*)d_in[2];
  const float* convb = (const float*)d_in[3];
  const float* bng   = (const float*)d_in[4];
  const float* bnb   = (const float*)d_in[5];
  float* out = (float*)d_out;

  char*   ws    = (char*)d_ws;
  float*  xmem  = (float*)ws;                                 // 8 MB
  float*  ybuf  = (float*)(ws + ((size_t)8 << 20));           // 8 MB
  double* acc   = (double*)(ws + ((size_t)16 << 20));         // 34 doubles
  float*  scsh  = (float*)(ws + ((size_t)16 << 20) + 1024);   // scale[16], shift[16]
  float*  e2tab = (float*)(ws + ((size_t)16 << 20) + 2048);   // |e_n|^2, 512 floats

  zero_accum_kernel<<<1, 64, 0, stream>>>(acc);
  e2_kernel<<<KCB / 64, 64, 0, stream>>>(emb, e2tab);
  memout_kernel<<<NPIX / 256, 256, 0, stream>>>(x, xmem);
  vq_conv_kernel<<<NTIL / 4, 128, 0, stream>>>(xmem, emb, e2tab, convw, convb, ybuf,
                                               acc, acc + 16, acc + 32);
  bn_final_kernel<<<1, 32, 0, stream>>>(acc, bng, bnb, scsh);
  lif_kernel<<<(NPIX * 16) / 256, 256, 0, stream>>>(ybuf, x, scsh, out, acc + 33);
  loss_kernel<<<1, 1, 0, stream>>>(acc, out + 33554432ull);
}